// Seq_Atten_82772609729104
// MI455X (gfx1250) — compile-verified
//
#include <hip/hip_runtime.h>
#include <hip/hip_bf16.h>

// ---------------------------------------------------------------------------
// Problem constants (from the reference)
// ---------------------------------------------------------------------------
constexpr int B   = 32;
constexpr int L   = 2048;
constexpr int C   = 1024;
constexpr int H   = 16;
constexpr int HD  = 64;     // C / H
constexpr int HID = 256;
constexpr int OUT = 128;

typedef __attribute__((ext_vector_type(2))) float v2f;
typedef __attribute__((ext_vector_type(8))) float v8f;

// ---------------------------------------------------------------------------
// CDNA5 async global->LDS copy (ASYNCcnt tracked), per CDNA5 ISA
// ---------------------------------------------------------------------------
__device__ __forceinline__ void async_copy_b128(const void* gaddr, unsigned lds_byte_off)
{
    asm volatile("global_load_async_to_lds_b128 %0, %1, off"
                 :: "v"(lds_byte_off), "v"(gaddr) : "memory");
}
__device__ __forceinline__ void wait_async0()
{
    asm volatile("s_wait_asynccnt 0x0" ::: "memory");
}
// Generic pointer -> wave-relative LDS byte offset (flat addr[31:0] == LDS offset)
__device__ __forceinline__ unsigned lds_off(const void* p)
{
    return (unsigned)(unsigned long long)p;
}

// ---------------------------------------------------------------------------
// Kernel A: q[b,:] = scale*(fvec_c[b] @ Wq^T + bq), qr likewise from dis_enc_c.
// Tiny (134 MFLOP). Wq (4 MB) stays L2-resident.
// ---------------------------------------------------------------------------
__global__ __launch_bounds__(256) void qproj_kernel(
    const float* __restrict__ fvec_c, const float* __restrict__ dis_enc_c,
    const float* __restrict__ W, const float* __restrict__ bias,
    float* __restrict__ q, float* __restrict__ qr)
{
    const int b     = blockIdx.x;
    const int which = blockIdx.y;                     // 0 -> q, 1 -> q_r
    const float* src = (which ? dis_enc_c : fvec_c) + (size_t)b * C;
    float*       dst = (which ? qr : q) + (size_t)b * C;
    const float scale = 0.125f;                       // hd^-0.5 = 1/8
    for (int e = threadIdx.x; e < C; e += 256) {
        const float* wrow = W + (size_t)e * C;        // Wq rows 0..C-1
        float acc = 0.f;
        for (int d = 0; d < C; d += 4) {
            float4 wv = *(const float4*)(wrow + d);
            float4 xv = *(const float4*)(src + d);
            acc = fmaf(wv.x, xv.x, acc); acc = fmaf(wv.y, xv.y, acc);
            acc = fmaf(wv.z, xv.z, acc); acc = fmaf(wv.w, xv.w, acc);
        }
        dst[e] = (acc + bias[e]) * scale;
    }
}

// ---------------------------------------------------------------------------
// Kernel B: fold Wk into the (length-1) queries.
//   pA[b,h,c] = sum_d (q+qr)[b,h*64+d] * Wk[h*64+d, c]
//   pB[b,h,c] = sum_d  q    [b,h*64+d] * Wk[h*64+d, c]
// Stored transposed, head-minor: pAT[b][c][h].
// ---------------------------------------------------------------------------
__global__ __launch_bounds__(256) void pmat_kernel(
    const float* __restrict__ W, const float* __restrict__ q,
    const float* __restrict__ qr, float* __restrict__ pAT, float* __restrict__ pBT)
{
    const int b = blockIdx.x;
    const int c = blockIdx.y * 16 + (threadIdx.x & 15);
    const int h = threadIdx.x >> 4;
    const float* wk = W + (size_t)C * C;              // Wk rows C..2C-1
    float pa = 0.f, pb = 0.f;
    for (int d = 0; d < HD; ++d) {
        float w   = wk[(size_t)(h * HD + d) * C + c];
        float qv  = q [b * C + h * HD + d];
        float qrv = qr[b * C + h * HD + d];
        pa = fmaf(qv + qrv, w, pa);
        pb = fmaf(qv,       w, pb);
    }
    pAT[((size_t)b * C + c) * H + h] = pa;
    pBT[((size_t)b * C + c) * H + h] = pb;
}

// ---------------------------------------------------------------------------
// Kernel C (dominant pass 1): logits[b,h,m] = X[b,m,:].pA[b,h,:]
//                                           + XR[b,m,:].pB[b,h,:]
// WMMA f32 16x16x4, one wave = one 16(m) x 16(h) tile; 8 waves/block cover
// 128 rows. Double-buffered async global->LDS staging of both the X tile and
// the PT (folded-query) chunk; inner loop is pure ds_load + v_wmma.
//
// LDS X-tile row stride = 36 floats (144 B): keeps async B128 stores 16 B
// aligned and the per-half-wave column reads on 16 distinct banks.
//
// Assumed f32 A/B fragment layout (mirrors ISA 16x4 A table):
//   vgpr0: K=0 (lanes 0-15), K=2 (lanes 16-31); vgpr1: K=1 / K=3.
// D layout per ISA: vgpr r: M=r (lanes 0-15), M=r+8 (lanes 16-31), N=lane&15.
// ---------------------------------------------------------------------------
constexpr int XS_STRIDE = 36;            // floats per staged row
constexpr int NCHUNK    = 2 * (C / 32);  // 2 phases x 32 K-chunks

__global__ __launch_bounds__(256) void logits_kernel(
    const float* __restrict__ X, const float* __restrict__ XR,
    const float* __restrict__ pAT, const float* __restrict__ pBT,
    float* __restrict__ logits)
{
    __shared__ float xs[2][128 * XS_STRIDE];   // 2 x 18 KB
    __shared__ float pts[2][32 * H];           // 2 x 2 KB

    const int b    = blockIdx.y;
    const int M0   = blockIdx.x * 128;
    const int tid  = threadIdx.x;
    const int wave = tid >> 5;
    const int lane = tid & 31;
    const int half = lane >> 4;
    const int lrow = lane & 15;
    const int wr   = wave * 16;

    const unsigned xs_base  = lds_off(&xs[0][0]);
    const unsigned pts_base = lds_off(&pts[0][0]);

    const float* srcX  = X  + ((size_t)b * L + M0) * C;
    const float* srcXR = XR + ((size_t)b * L + M0) * C;
    const float* ptA   = pAT + (size_t)b * C * H;
    const float* ptB   = pBT + (size_t)b * C * H;

    // Issue async stage of one chunk (ch in [0, NCHUNK)).
    auto issue_chunk = [&](int ch) {
        const int ph  = ch >> 5;                 // 0: X/pA, 1: XR/pB
        const int k0  = (ch & 31) * 32;
        const int buf = ch & 1;
        const float* srcb = (ph ? srcXR : srcX);
        const float* PT   = (ph ? ptB : ptA);
        #pragma unroll
        for (int i = 0; i < 4; ++i) {            // 128 rows x 32 floats = 1024 B128
            int idx = tid + i * 256;
            int row = idx >> 3;                  // 8 x B128 per row
            int c4  = (idx & 7) << 2;
            async_copy_b128(srcb + (size_t)row * C + k0 + c4,
                            xs_base + (unsigned)(buf * 128 * XS_STRIDE + row * XS_STRIDE + c4) * 4u);
        }
        if (tid < 128)                           // 32 x 16 floats = 128 B128
            async_copy_b128(PT + (size_t)k0 * H + tid * 4,
                            pts_base + (unsigned)(buf * 32 * H + tid * 4) * 4u);
    };

    v8f acc = {};
    issue_chunk(0);
    for (int ch = 0; ch < NCHUNK; ++ch) {
        const int buf = ch & 1;
        wait_async0();           // this wave's stage of chunk `ch` is in LDS
        __syncthreads();         // everyone's stage done; prev buffer free
        if (ch + 1 < NCHUNK) issue_chunk(ch + 1);

        const float* xrow = &xs[buf][(wr + lrow) * XS_STRIDE];
        const float* pbuf = &pts[buf][0];
        #pragma unroll
        for (int kk = 0; kk < 8; ++kk) {
            const int kb = kk * 4 + half * 2;
            v2f a, bf;
            a.x  = xrow[kb + 0];
            a.y  = xrow[kb + 1];
            bf.x = pbuf[(kb + 0) * H + lrow];    // B[k][n] = P^T[k0+k][h=n]
            bf.y = pbuf[(kb + 1) * H + lrow];
            acc = __builtin_amdgcn_wmma_f32_16x16x4_f32(
                false, a, false, bf, (short)0, acc, false, false);
        }
    }
    // Per-(b,h) additive constants (bias dot terms) are softmax-invariant: dropped.
    const int h     = lrow;
    const int mbase = M0 + wr + half * 8;
    #pragma unroll
    for (int r = 0; r < 8; ++r)
        logits[((size_t)b * H + h) * L + (mbase + r)] = acc[r];
}

// ---------------------------------------------------------------------------
// Kernel: softmax over m (in place), one block per (b,h) row of 2048.
// ---------------------------------------------------------------------------
__global__ __launch_bounds__(256) void softmax_kernel(float* __restrict__ logits)
{
    __shared__ float red[256];
    float* p = logits + (size_t)blockIdx.x * L;
    const int t = threadIdx.x;
    float mx = -3.402823466e38f;
    for (int i = t; i < L; i += 256) mx = fmaxf(mx, p[i]);
    red[t] = mx; __syncthreads();
    for (int s = 128; s > 0; s >>= 1) { if (t < s) red[t] = fmaxf(red[t], red[t + s]); __syncthreads(); }
    mx = red[0]; __syncthreads();
    float sum = 0.f;
    for (int i = t; i < L; i += 256) { float e = __expf(p[i] - mx); p[i] = e; sum += e; }
    red[t] = sum; __syncthreads();
    for (int s = 128; s > 0; s >>= 1) { if (t < s) red[t] += red[t + s]; __syncthreads(); }
    const float inv = 1.f / red[0];
    for (int i = t; i < L; i += 256) p[i] *= inv;
}

// ---------------------------------------------------------------------------
// Kernel (dominant pass 2): S[b,h,c] = sum_m a[b,h,m] * X[b,m,c].
// Thread owns one column c for all 16 heads; attention weights broadcast from
// LDS (all lanes read same address -> no conflicts). Streams X once more.
// ---------------------------------------------------------------------------
__global__ __launch_bounds__(256) void wsum_kernel(
    const float* __restrict__ X, const float* __restrict__ A, float* __restrict__ S)
{
    __shared__ float as[H][128];
    const int b = blockIdx.x;
    const int c = blockIdx.y * 256 + threadIdx.x;
    const int t = threadIdx.x;
    float acc[H];
    #pragma unroll
    for (int h = 0; h < H; ++h) acc[h] = 0.f;
    for (int m0 = 0; m0 < L; m0 += 128) {
        __syncthreads();
        #pragma unroll
        for (int i = 0; i < 8; ++i) {              // 16x128 weights, coalesced
            int idx = t + i * 256;
            int h = idx >> 7, mm = idx & 127;
            as[h][mm] = A[((size_t)b * H + h) * L + m0 + mm];
        }
        __syncthreads();
        for (int mm = 0; mm < 128; ++mm) {
            float x = X[((size_t)b * L + m0 + mm) * C + c];
            #pragma unroll
            for (int h = 0; h < H; ++h) acc[h] = fmaf(as[h][mm], x, acc[h]);
        }
    }
    #pragma unroll
    for (int h = 0; h < H; ++h)
        S[((size_t)b * H + h) * C + c] = acc[h];
}

// ---------------------------------------------------------------------------
// Kernel: V projection moved after the reduction (sum a = 1 keeps bias exact):
//   VO[b, h*64+c'] = Wv[h*64+c', :] . S[b,h,:] + bv[h*64+c']
// ---------------------------------------------------------------------------
__global__ __launch_bounds__(256) void vproj_kernel(
    const float* __restrict__ W, const float* __restrict__ bias,
    const float* __restrict__ S, float* __restrict__ VO)
{
    const int b = blockIdx.x;
    for (int e = threadIdx.x; e < C; e += 256) {
        const float* wrow = W + (size_t)(2 * C + e) * C;   // Wv rows 2C..3C-1
        const float* srow = S + ((size_t)b * H + (e >> 6)) * C;
        float acc = bias[2 * C + e];
        for (int d = 0; d < C; d += 4) {
            float4 wv = *(const float4*)(wrow + d);
            float4 sv = *(const float4*)(srow + d);
            acc = fmaf(wv.x, sv.x, acc); acc = fmaf(wv.y, sv.y, acc);
            acc = fmaf(wv.z, sv.z, acc); acc = fmaf(wv.w, sv.w, acc);
        }
        VO[(size_t)b * C + e] = acc;
    }
}

// ---------------------------------------------------------------------------
// Generic small GEMV: out[b,e] = W[e,:K].in[b,:K] + bias[e]   (N,K small)
// ---------------------------------------------------------------------------
__global__ __launch_bounds__(256) void gemv_kernel(
    const float* __restrict__ W, const float* __restrict__ bias,
    const float* __restrict__ in, float* __restrict__ out, int N, int K)
{
    const int b = blockIdx.x;
    const float* src = in + (size_t)b * K;
    for (int e = threadIdx.x; e < N; e += blockDim.x) {
        const float* wrow = W + (size_t)e * K;
        float acc = bias[e];
        for (int d = 0; d < K; d += 4) {
            float4 wv = *(const float4*)(wrow + d);
            float4 xv = *(const float4*)(src + d);
            acc = fmaf(wv.x, xv.x, acc); acc = fmaf(wv.y, xv.y, acc);
            acc = fmaf(wv.z, xv.z, acc); acc = fmaf(wv.w, xv.w, acc);
        }
        out[(size_t)b * N + e] = acc;
    }
}

// ---------------------------------------------------------------------------
// Kernel: LayerNorm over the last dim (one block per batch row).
// ---------------------------------------------------------------------------
__global__ __launch_bounds__(256) void ln_kernel(
    const float* __restrict__ in, const float* __restrict__ g,
    const float* __restrict__ bta, float* __restrict__ out, int N)
{
    __shared__ float red[256];
    const float* p = in + (size_t)blockIdx.x * N;
    const int t = threadIdx.x;
    float sum = 0.f;
    for (int i = t; i < N; i += 256) sum += p[i];
    red[t] = sum; __syncthreads();
    for (int s = 128; s > 0; s >>= 1) { if (t < s) red[t] += red[t + s]; __syncthreads(); }
    const float mean = red[0] / N; __syncthreads();
    float vs = 0.f;
    for (int i = t; i < N; i += 256) { float d = p[i] - mean; vs = fmaf(d, d, vs); }
    red[t] = vs; __syncthreads();
    for (int s = 128; s > 0; s >>= 1) { if (t < s) red[t] += red[t + s]; __syncthreads(); }
    const float rstd = rsqrtf(red[0] / N + 1e-5f);
    for (int i = t; i < N; i += 256)
        out[(size_t)blockIdx.x * N + i] = (p[i] - mean) * rstd * g[i] + bta[i];
}

// ---------------------------------------------------------------------------
// Host-side launch
// ---------------------------------------------------------------------------
extern "C" void kernel_launch(void* const* d_in, const int* in_sizes, int n_in,
                              void* d_out, int out_size, void* d_ws, size_t ws_size,
                              hipStream_t stream)
{
    const float* fvec_c      = (const float*)d_in[0];
    const float* fvec_seq    = (const float*)d_in[1];
    const float* dis_enc_c   = (const float*)d_in[2];
    const float* dis_enc_seq = (const float*)d_in[3];
    const float* in_proj_w   = (const float*)d_in[4];
    const float* in_proj_b   = (const float*)d_in[5];
    const float* out_w       = (const float*)d_in[6];
    const float* out_b       = (const float*)d_in[7];
    const float* ln1_g       = (const float*)d_in[8];
    const float* ln1_b       = (const float*)d_in[9];
    const float* w1          = (const float*)d_in[10];
    const float* b1          = (const float*)d_in[11];
    const float* ln2_g       = (const float*)d_in[12];
    const float* ln2_b       = (const float*)d_in[13];
    const float* w2          = (const float*)d_in[14];
    const float* b2          = (const float*)d_in[15];

    float* ws     = (float*)d_ws;
    float* q      = ws;                    // B*C
    float* qr     = q      + B * C;        // B*C
    float* pAT    = qr     + B * C;        // B*C*H
    float* pBT    = pAT    + (size_t)B * C * H;
    float* logits = pBT    + (size_t)B * C * H;   // B*H*L (becomes softmax a)
    float* S      = logits + (size_t)B * H * L;   // B*H*C
    float* vo     = S      + (size_t)B * H * C;   // B*C
    float* o1     = vo     + B * C;        // B*C
    float* x1     = o1     + B * C;        // B*C
    float* h1     = x1     + B * C;        // B*HID
    float* h2     = h1     + B * HID;      // B*HID

    qproj_kernel  <<<dim3(B, 2),      256, 0, stream>>>(fvec_c, dis_enc_c, in_proj_w, in_proj_b, q, qr);
    pmat_kernel   <<<dim3(B, C / 16), 256, 0, stream>>>(in_proj_w, q, qr, pAT, pBT);
    logits_kernel <<<dim3(L / 128, B),256, 0, stream>>>(fvec_seq, dis_enc_seq, pAT, pBT, logits);
    softmax_kernel<<<B * H,           256, 0, stream>>>(logits);
    wsum_kernel   <<<dim3(B, C / 256),256, 0, stream>>>(fvec_seq, logits, S);
    vproj_kernel  <<<B,               256, 0, stream>>>(in_proj_w, in_proj_b, S, vo);
    gemv_kernel   <<<B,               256, 0, stream>>>(out_w, out_b, vo, o1, C, C);
    ln_kernel     <<<B,               256, 0, stream>>>(o1, ln1_g, ln1_b, x1, C);
    gemv_kernel   <<<B,               256, 0, stream>>>(w1, b1, x1, h1, HID, C);
    ln_kernel     <<<B,               256, 0, stream>>>(h1, ln2_g, ln2_b, h2, HID);
    gemv_kernel   <<<B,               256, 0, stream>>>(w2, b2, h2, (float*)d_out, OUT, HID);
}